// ViewSignature_4157528342910
// MI455X (gfx1250) — compile-verified
//
#include <hip/hip_runtime.h>

// V_WMMA_F32_16X16X4_F32 operand vectors (wave32):
//   A 16x4 f32  -> 2 VGPRs/lane, B 4x16 f32 -> 2 VGPRs/lane, C/D 16x16 -> 8 VGPRs/lane
typedef __attribute__((ext_vector_type(2))) float v2f;
typedef __attribute__((ext_vector_type(8))) float v8f;

#define NCH 10           // channels
#define PLEN 64          // path length
#define NSTEP 64         // 63 real increments + 1 zero pad (algebraic no-op)
#define NREAL 63
#define NROW 1000        // valid rows of level-4 matrix view (flattened first 3 indices)
#define NROW_PAD 1024    // padded to 64 M-tiles -> branchless 8 tiles per wave
#define NGRP 16          // 64 steps / 4 (K of the f32 WMMA)
#define OUT_STRIDE 11110 // 10 + 100 + 1000 + 10000

// One workgroup (256 threads = 8 waves) per batch element.
// Horner/Chen step with increment dx (S_k are running signature levels):
//   T1 = S1 + dx/2                          ; S2 += T1 (x) dx
//   T2 = S2 + (S1/2 + dx/6)(x)dx            ; S3 += T2 (x) dx
//   T3 = S3 + (S2/2 + (S1/6 + dx/24)(x)dx)(x)dx
//   S4 += T3 (x) dx                         ; S1 += dx
// The S4 update over a group of 4 steps is a 1024x4 * 4x16 GEMM (valid 1000x10)
// -> v_wmma_f32_16x16x4_f32, with S4 resident in WMMA accumulator VGPRs all scan long.
__global__ __launch_bounds__(256) void sig_scan_wmma_kernel(const float* __restrict__ x,
                                                            float* __restrict__ out,
                                                            int batch)
{
    __shared__ float dxs[NSTEP][16];       // increments; cols 10..15 and step 63 are zero
    __shared__ float S1[NCH];
    __shared__ float S2[100];
    __shared__ float S3[NROW];
    __shared__ float T1s[NCH];
    __shared__ float T2s[100];
    __shared__ float V2s[100];
    __shared__ float Astage[NROW_PAD][4];  // T3 columns for one 4-step group (rows 1000..1023 = 0)

    const int tid = threadIdx.x;
    const int b   = blockIdx.x;
    if (b >= batch) return;
    const float* xb = x + (size_t)b * (NCH * PLEN);
    float* ob = out + (size_t)b * OUT_STRIDE;

    // ---- init: increments (zero-padded in one pass) + zero state ----
    for (int i = tid; i < NSTEP * 16; i += 256) {
        const int t = i >> 4, c = i & 15;
        float v = 0.f;
        if (t < NREAL && c < NCH) v = xb[c * PLEN + t + 1] - xb[c * PLEN + t];
        dxs[t][c] = v;
    }
    if (tid < NCH) S1[tid] = 0.f;
    for (int i = tid; i < 100; i += 256) S2[i] = 0.f;
    for (int i = tid; i < NROW; i += 256) S3[i] = 0.f;
    for (int i = tid; i < (NROW_PAD - NROW) * 4; i += 256)    // pad rows stay zero
        Astage[NROW + (i >> 2)][i & 3] = 0.f;
    __syncthreads();

    const int wave  = tid >> 5;          // 0..7 (uniform per wave)
    const int lane  = tid & 31;
    const int nlo   = lane & 15;         // tile column / A-row-within-tile
    const int khalf = (lane >> 4) << 1;  // K pair selector: 0 (lanes 0-15) or 2 (lanes 16-31)

    v8f acc[8];
    #pragma unroll
    for (int i = 0; i < 8; ++i) acc[i] = (v8f){0.f,0.f,0.f,0.f,0.f,0.f,0.f,0.f};

    for (int g = 0; g < NGRP; ++g) {
        // ---- scalar scan of 4 steps: levels 1-3 + stage T3 columns ----
        #pragma unroll 1
        for (int k = 0; k < 4; ++k) {
            const float* dx = dxs[4 * g + k];
            // phase 1: temporaries from OLD state
            if (tid < 100) {
                const int i0 = tid / 10, j = tid % 10;
                const float dj = dx[j];
                V2s[tid] = S2[tid] * 0.5f + (S1[i0] * (1.f/6.f) + dx[i0] * (1.f/24.f)) * dj;
                T2s[tid] = S2[tid]        + (S1[i0] * 0.5f      + dx[i0] * (1.f/6.f )) * dj;
            }
            if (tid >= 128 && tid < 128 + NCH) {
                const int i0 = tid - 128;
                T1s[i0] = S1[i0] + dx[i0] * 0.5f;
            }
            __syncthreads();
            // phase 2: write T3 column, update S3/S2/S1 (each element owned by one thread)
            for (int e = tid; e < NROW; e += 256) {
                const int rq = e / 10, j = e % 10;
                const float dj = dx[j];
                const float s3old = S3[e];
                Astage[e][k] = s3old + V2s[rq] * dj;   // T3
                S3[e]        = s3old + T2s[rq] * dj;
            }
            if (tid < 100) S2[tid] += T1s[tid / 10] * dx[tid % 10];
            if (tid >= 128 && tid < 128 + NCH) S1[tid - 128] += dx[tid - 128];
            __syncthreads();
        }

        // ---- matrix phase: S4 += Astage[1024x4] * dx[4x16] via v_wmma_f32_16x16x4_f32 ----
        // B fragment: rows K=khalf,khalf+1 of the 4x16 dx block (cols 10..15 are zero-padded).
        const int t0 = 4 * g;
        v2f bfrag;
        bfrag.x = dxs[t0 + khalf][nlo];
        bfrag.y = dxs[t0 + khalf + 1][nlo];
        // Each wave owns 8 contiguous M-tiles: rows [128*wave, 128*wave+128). Branchless.
        const int rbase = 128 * wave + nlo;
        #pragma unroll
        for (int i = 0; i < 8; ++i) {
            const int r = rbase + 16 * i;
            v2f afrag;
            afrag.x = Astage[r][khalf];
            afrag.y = Astage[r][khalf + 1];
            acc[i] = __builtin_amdgcn_wmma_f32_16x16x4_f32(
                /*neg_a=*/false, afrag, /*neg_b=*/false, bfrag,
                /*c_mod=*/(short)0, acc[i], /*reuse_a=*/false, /*reuse_b=*/false);
        }
        __syncthreads();   // Astage is rewritten by the next group's scan
    }

    // ---- writeback ----
    if (tid < NCH) ob[tid] = S1[tid];
    if (tid < 100) ob[10 + tid] = S2[tid];
    for (int e = tid; e < NROW; e += 256) ob[110 + e] = S3[e];

    // S4: D[m][n] of tile mt=8*wave+i -> out row r = 16*mt + m (m = vgpr + 8*(lane>=16)),
    // col n = lane&15; valid region r < 1000, n < 10.
    const int mhi = (lane >> 4) << 3;
    #pragma unroll
    for (int i = 0; i < 8; ++i) {
        const int rt = 128 * wave + 16 * i + mhi;
        if (nlo < NCH) {
            #pragma unroll
            for (int v = 0; v < 8; ++v) {
                const int r = rt + v;
                if (r < NROW) ob[1110 + r * NCH + nlo] = acc[i][v];
            }
        }
    }
}

extern "C" void kernel_launch(void* const* d_in, const int* in_sizes, int n_in,
                              void* d_out, int out_size, void* d_ws, size_t ws_size,
                              hipStream_t stream) {
    const float* x = (const float*)d_in[0];
    float* out = (float*)d_out;
    const int batch = in_sizes[0] / (NCH * PLEN);   // 2048
    sig_scan_wmma_kernel<<<batch, 256, 0, stream>>>(x, out, batch);
}